// ClusterGAT_61495341744282
// MI455X (gfx1250) — compile-verified
//
#include <hip/hip_runtime.h>
#include <math.h>

typedef float v2f __attribute__((ext_vector_type(2)));
typedef float v8f __attribute__((ext_vector_type(8)));

#define NEG_INF_KEY ((int)0x807FFFFF)   // key(-inf)

__device__ __forceinline__ int fkey(float f) {
  int b = __float_as_int(f);
  return b >= 0 ? b : (b ^ 0x7FFFFFFF);
}
__device__ __forceinline__ float funkey(int k) {
  return __int_as_float(k >= 0 ? k : (k ^ 0x7FFFFFFF));
}
__device__ __forceinline__ float leaky(float v) { return v > 0.f ? v : 0.2f * v; }

// ---------------- init helpers ----------------
__global__ void k_zero_f(float* __restrict__ p, int n) {
  int i = blockIdx.x * blockDim.x + threadIdx.x;
  if (i < n) p[i] = 0.f;
}
__global__ void k_set_i(int* __restrict__ p, int val, int n) {
  int i = blockIdx.x * blockDim.x + threadIdx.x;
  if (i < n) p[i] = val;
}
// W2 [128,7] -> zero-padded [128,16]
__global__ void k_pad_w2(const float* __restrict__ W2, float* __restrict__ W2p) {
  int i = blockIdx.x * blockDim.x + threadIdx.x;
  if (i >= 128 * 16) return;
  int row = i >> 4, col = i & 15;
  W2p[i] = (col < 7) ? W2[row * 7 + col] : 0.f;
}

// ---------------- layer-1 GEMM: h = x @ W1, fp32 WMMA 16x16x4 ----------------
// block = 256 threads (8 waves). wave w -> cols [16w,16w+16); block -> 16 rows.
__global__ void k_gemm1(const float* __restrict__ x, const float* __restrict__ W,
                        float* __restrict__ h, int N) {
  const int wave = threadIdx.x >> 5;
  const int lane = threadIdx.x & 31;
  const int row0 = blockIdx.x << 4;
  const int col0 = wave << 4;
  const int m    = lane & 15;
  const int kh   = (lane >> 4) << 1;          // 0 or 2 (K half-split)
  int r = row0 + m; if (r > N - 1) r = N - 1; // clamp loads, keep EXEC full for WMMA
  const float* __restrict__ xr = x + (size_t)r * 128;
  v8f acc = {};
#pragma unroll
  for (int k = 0; k < 128; k += 4) {
    v2f a, b;
    a.x = xr[k + kh];                                  // A 16x4: V0 = K0/K2, V1 = K1/K3
    a.y = xr[k + kh + 1];
    b.x = W[(size_t)(k + kh) * 128 + col0 + m];        // B 4x16: row striped across lanes
    b.y = W[(size_t)(k + kh + 1) * 128 + col0 + m];
    acc = __builtin_amdgcn_wmma_f32_16x16x4_f32(false, a, false, b,
                                                (short)0, acc, false, false);
  }
  const int rbase = row0 + ((lane >> 4) << 3);         // D: VGPR j -> row j or j+8
  const int c = col0 + m;
#pragma unroll
  for (int j = 0; j < 8; ++j) {
    int rr = rbase + j;
    if (rr < N) h[(size_t)rr * 128 + c] = acc[j];
  }
}

// ---------------- layer-1 attention coefficients ----------------
__global__ void k_alpha1(const float* __restrict__ h, const float* __restrict__ a_src,
                         const float* __restrict__ a_dst, float* __restrict__ as,
                         float* __restrict__ ad, int N) {
  int i = blockIdx.x * blockDim.x + threadIdx.x;      // i over N*2 (node,head)
  if (i >= N * 2) return;
  int n = i >> 1, head = i & 1;
  const float* __restrict__ hp = h + (size_t)n * 128 + head * 64;
  const float* __restrict__ av = a_src + head * 64;
  const float* __restrict__ dv = a_dst + head * 64;
  float ss = 0.f, sd = 0.f;
#pragma unroll 8
  for (int c = 0; c < 64; ++c) { ss += hp[c] * av[c]; sd += hp[c] * dv[c]; }
  as[i] = ss; ad[i] = sd;
}

// ---------------- edge passes, layer 1 (H=2) ----------------
__device__ __forceinline__ void edge_sd(const long long* __restrict__ ei, int e, int E,
                                        int& s, int& d) {
  if (e < E) { s = (int)ei[e]; d = (int)ei[(size_t)E + e]; }
  else       { s = d = e - E; }
}

__global__ void k_edge_max1(const long long* __restrict__ ei, const float* __restrict__ as,
                            const float* __restrict__ ad, int* __restrict__ mk,
                            int E, int N) {
  int e = blockIdx.x * blockDim.x + threadIdx.x;
  if (e >= E + N) return;
  int s, d; edge_sd(ei, e, E, s, d);
#pragma unroll
  for (int head = 0; head < 2; ++head) {
    float v = leaky(as[s * 2 + head] + ad[d * 2 + head]);
    atomicMax(&mk[d * 2 + head], fkey(v));
  }
}

__global__ void k_edge_sum1(const long long* __restrict__ ei, const float* __restrict__ as,
                            const float* __restrict__ ad, const int* __restrict__ mk,
                            float* __restrict__ ssum, int E, int N) {
  int e = blockIdx.x * blockDim.x + threadIdx.x;
  if (e >= E + N) return;
  int s, d; edge_sd(ei, e, E, s, d);
#pragma unroll
  for (int head = 0; head < 2; ++head) {
    float v = leaky(as[s * 2 + head] + ad[d * 2 + head]);
    atomicAdd(&ssum[d * 2 + head], __expf(v - funkey(mk[d * 2 + head])));
  }
}

// one wave per edge: lane covers 4 consecutive feature floats (head = lane>>4)
__global__ void k_edge_agg1(const long long* __restrict__ ei, const float* __restrict__ as,
                            const float* __restrict__ ad, const int* __restrict__ mk,
                            const float* __restrict__ ssum, const float* __restrict__ h,
                            float* __restrict__ out, int E, int N) {
  const int wave = threadIdx.x >> 5;
  const int lane = threadIdx.x & 31;
  const int e = blockIdx.x * 8 + wave;
  if (e >= E + N) return;
  int s, d; edge_sd(ei, e, E, s, d);
  const int head = lane >> 4;
  const int idx  = d * 2 + head;
  float v     = leaky(as[s * 2 + head] + ad[idx]);
  float alpha = __expf(v - funkey(mk[idx])) / (ssum[idx] + 1e-16f);
  const float4 hv = *reinterpret_cast<const float4*>(h + (size_t)s * 128 + lane * 4);
  float* op = out + (size_t)d * 128 + lane * 4;
  atomicAdd(op + 0, hv.x * alpha);
  atomicAdd(op + 1, hv.y * alpha);
  atomicAdd(op + 2, hv.z * alpha);
  atomicAdd(op + 3, hv.w * alpha);
}

__global__ void k_bias_relu(float* __restrict__ p, const float* __restrict__ b, int N) {
  int i = blockIdx.x * blockDim.x + threadIdx.x;
  if (i >= N * 128) return;
  float v = p[i] + b[i & 127];
  p[i] = v > 0.f ? v : 0.f;
}

// ---------------- layer-2 GEMM: z2 = h2 @ W2pad (128x16), keep 7 cols ----------------
__global__ void k_gemm2(const float* __restrict__ h, const float* __restrict__ Wp,
                        float* __restrict__ z, int N) {
  const int wave = threadIdx.x >> 5;
  const int lane = threadIdx.x & 31;
  const int row0 = blockIdx.x * 128 + (wave << 4);
  const int m    = lane & 15;
  const int kh   = (lane >> 4) << 1;
  int r = row0 + m; if (r > N - 1) r = N - 1;
  const float* __restrict__ hr = h + (size_t)r * 128;
  v8f acc = {};
#pragma unroll
  for (int k = 0; k < 128; k += 4) {
    v2f a, b;
    a.x = hr[k + kh];
    a.y = hr[k + kh + 1];
    b.x = Wp[(k + kh) * 16 + m];
    b.y = Wp[(k + kh + 1) * 16 + m];
    acc = __builtin_amdgcn_wmma_f32_16x16x4_f32(false, a, false, b,
                                                (short)0, acc, false, false);
  }
  const int rbase = row0 + ((lane >> 4) << 3);
  if (m < 7) {
#pragma unroll
    for (int j = 0; j < 8; ++j) {
      int rr = rbase + j;
      if (rr < N) z[(size_t)rr * 7 + m] = acc[j];
    }
  }
}

__global__ void k_alpha2(const float* __restrict__ z, const float* __restrict__ a_src,
                         const float* __restrict__ a_dst, float* __restrict__ as,
                         float* __restrict__ ad, int N) {
  int n = blockIdx.x * blockDim.x + threadIdx.x;
  if (n >= N) return;
  float ss = 0.f, sd = 0.f;
#pragma unroll
  for (int c = 0; c < 7; ++c) {
    float v = z[(size_t)n * 7 + c];
    ss += v * a_src[c]; sd += v * a_dst[c];
  }
  as[n] = ss; ad[n] = sd;
}

__global__ void k_edge_max2(const long long* __restrict__ ei, const float* __restrict__ as,
                            const float* __restrict__ ad, int* __restrict__ mk,
                            int E, int N) {
  int e = blockIdx.x * blockDim.x + threadIdx.x;
  if (e >= E + N) return;
  int s, d; edge_sd(ei, e, E, s, d);
  atomicMax(&mk[d], fkey(leaky(as[s] + ad[d])));
}

__global__ void k_edge_sum2(const long long* __restrict__ ei, const float* __restrict__ as,
                            const float* __restrict__ ad, const int* __restrict__ mk,
                            float* __restrict__ ssum, int E, int N) {
  int e = blockIdx.x * blockDim.x + threadIdx.x;
  if (e >= E + N) return;
  int s, d; edge_sd(ei, e, E, s, d);
  float v = leaky(as[s] + ad[d]);
  atomicAdd(&ssum[d], __expf(v - funkey(mk[d])));
}

// 8 threads per edge (c = tid&7, c<7 active)
__global__ void k_edge_agg2(const long long* __restrict__ ei, const float* __restrict__ as,
                            const float* __restrict__ ad, const int* __restrict__ mk,
                            const float* __restrict__ ssum, const float* __restrict__ z,
                            float* __restrict__ out, int E, int N) {
  int i = blockIdx.x * blockDim.x + threadIdx.x;
  int e = i >> 3, c = i & 7;
  if (e >= E + N || c >= 7) return;
  int s, d; edge_sd(ei, e, E, s, d);
  float v     = leaky(as[s] + ad[d]);
  float alpha = __expf(v - funkey(mk[d])) / (ssum[d] + 1e-16f);
  atomicAdd(&out[(size_t)d * 7 + c], z[(size_t)s * 7 + c] * alpha);
}

__global__ void k_logsoftmax(const float* __restrict__ acc, const float* __restrict__ b2,
                             float* __restrict__ out, int N) {
  int n = blockIdx.x * blockDim.x + threadIdx.x;
  if (n >= N) return;
  float v[7];
  float vmax = -INFINITY;
#pragma unroll
  for (int c = 0; c < 7; ++c) { v[c] = acc[(size_t)n * 7 + c] + b2[c]; vmax = fmaxf(vmax, v[c]); }
  float se = 0.f;
#pragma unroll
  for (int c = 0; c < 7; ++c) se += __expf(v[c] - vmax);
  float lse = vmax + __logf(se);
#pragma unroll
  for (int c = 0; c < 7; ++c) out[(size_t)n * 7 + c] = v[c] - lse;
}

// ---------------- launch ----------------
extern "C" void kernel_launch(void* const* d_in, const int* in_sizes, int n_in,
                              void* d_out, int out_size, void* d_ws, size_t ws_size,
                              hipStream_t stream) {
  const float*     x      = (const float*)d_in[0];
  const long long* ei     = (const long long*)d_in[1];
  const float*     W1     = (const float*)d_in[2];
  const float*     a_src1 = (const float*)d_in[3];
  const float*     a_dst1 = (const float*)d_in[4];
  const float*     b1     = (const float*)d_in[5];
  const float*     W2     = (const float*)d_in[6];
  const float*     a_src2 = (const float*)d_in[7];
  const float*     a_dst2 = (const float*)d_in[8];
  const float*     b2     = (const float*)d_in[9];
  float* out = (float*)d_out;

  const int N = in_sizes[0] / 128;   // 100000
  const int E = in_sizes[1] / 2;     // 1600000
  const int Etot = E + N;

  // workspace carve-up (all region sizes are multiples of 16 floats)
  float* ws = (float*)d_ws;  size_t off = 0;
  float* h1   = ws + off; off += (size_t)N * 128;
  float* out1 = ws + off; off += (size_t)N * 128;
  float* as1  = ws + off; off += (size_t)N * 2;
  float* ad1  = ws + off; off += (size_t)N * 2;
  float* s1   = ws + off; off += (size_t)N * 2;
  int*   m1   = (int*)(ws + off); off += (size_t)N * 2;
  float* z2   = ws + off; off += (size_t)N * 7 + 16;
  float* as2  = ws + off; off += (size_t)N;
  float* ad2  = ws + off; off += (size_t)N;
  float* s2   = ws + off; off += (size_t)N;
  int*   m2   = (int*)(ws + off); off += (size_t)N;
  float* out2 = ws + off; off += (size_t)N * 7 + 16;
  float* W2p  = ws + off; off += 128 * 16;
  (void)ws_size; (void)n_in; (void)out_size;

  const int T = 256;
  auto nb = [](int n, int t) { return (n + t - 1) / t; };

  // ---- init (re-run every call: ws is poisoned and never restored) ----
  k_zero_f<<<nb(N * 128, T), T, 0, stream>>>(out1, N * 128);
  k_zero_f<<<nb(N * 2,   T), T, 0, stream>>>(s1, N * 2);
  k_set_i <<<nb(N * 2,   T), T, 0, stream>>>(m1, NEG_INF_KEY, N * 2);
  k_zero_f<<<nb(N * 7,   T), T, 0, stream>>>(out2, N * 7);
  k_zero_f<<<nb(N,       T), T, 0, stream>>>(s2, N);
  k_set_i <<<nb(N,       T), T, 0, stream>>>(m2, NEG_INF_KEY, N);
  k_pad_w2<<<nb(128 * 16, T), T, 0, stream>>>(W2, W2p);

  // ---- layer 1 ----
  k_gemm1<<<nb(N, 16), 256, 0, stream>>>(x, W1, h1, N);
  k_alpha1<<<nb(N * 2, T), T, 0, stream>>>(h1, a_src1, a_dst1, as1, ad1, N);
  k_edge_max1<<<nb(Etot, T), T, 0, stream>>>(ei, as1, ad1, m1, E, N);
  k_edge_sum1<<<nb(Etot, T), T, 0, stream>>>(ei, as1, ad1, m1, s1, E, N);
  k_edge_agg1<<<nb(Etot, 8), 256, 0, stream>>>(ei, as1, ad1, m1, s1, h1, out1, E, N);
  k_bias_relu<<<nb(N * 128, T), T, 0, stream>>>(out1, b1, N);

  // ---- layer 2 ----
  k_gemm2<<<nb(N, 128), 256, 0, stream>>>(out1, W2p, z2, N);
  k_alpha2<<<nb(N, T), T, 0, stream>>>(z2, a_src2, a_dst2, as2, ad2, N);
  k_edge_max2<<<nb(Etot, T), T, 0, stream>>>(ei, as2, ad2, m2, E, N);
  k_edge_sum2<<<nb(Etot, T), T, 0, stream>>>(ei, as2, ad2, m2, s2, E, N);
  k_edge_agg2<<<nb(Etot * 8, T), T, 0, stream>>>(ei, as2, ad2, m2, s2, z2, out2, E, N);
  k_logsoftmax<<<nb(N, T), T, 0, stream>>>(out2, b2, out, N);
}